// SIRConvBase_2645699854683
// MI455X (gfx1250) — compile-verified
//
#include <hip/hip_runtime.h>

// ---------------------------------------------------------------------------
// SIR graph conv, restructured:
//   out[v] = norm[v] * ( S[v]*(feat[v] @ W1^T + b) + agg[v] @ W2^T )
// where S[v] = sum_{e:dst=v} norm[src_e], agg[v] = sum norm[src_e]*feat[src_e]
// Single fused GEMM: X = [S*feat | agg]  (N x 256)  times  W^T (256 x 128)
// GEMM runs on V_WMMA_F32_16X16X4_F32 (fp32 WMMA, wave32).
// ---------------------------------------------------------------------------

typedef float v2f __attribute__((ext_vector_type(2)));
typedef float v8f __attribute__((ext_vector_type(8)));

#define DF 128
#define XS_STRIDE 260   // 16-row LDS tile, padded stride (260%64==4) to spread banks

__global__ void zero_ws_kernel(float* __restrict__ p, int n) {
    int i = blockIdx.x * blockDim.x + threadIdx.x;
    if (i < n) p[i] = 0.0f;
}

__global__ void deg_kernel(const int* __restrict__ dst, float* __restrict__ deg, int E) {
    int e = blockIdx.x * blockDim.x + threadIdx.x;
    if (e < E) atomicAdd(&deg[dst[e]], 1.0f);
}

__global__ void norm_kernel(float* __restrict__ deg_norm, int N) {
    int v = blockIdx.x * blockDim.x + threadIdx.x;
    if (v < N) {
        float d = deg_norm[v];
        deg_norm[v] = rsqrtf(fmaxf(d, 1.0f));   // deg array becomes norm in place
    }
}

// One wave32 per edge: lane k handles features [4k, 4k+4).
__global__ void __launch_bounds__(256)
scatter_kernel(const float* __restrict__ feat,
               const int*   __restrict__ src,
               const int*   __restrict__ dst,
               const float* __restrict__ norm,
               float* __restrict__ S,
               float* __restrict__ agg,
               int E) {
    const int lane = threadIdx.x & 31;
    const int wv   = threadIdx.x >> 5;
    const int e    = blockIdx.x * 8 + wv;
    if (e >= E) return;
    const int s = src[e];
    const int d = dst[e];
    const float w = norm[s];
    if (lane == 0) atomicAdd(&S[d], w);
    const float4 f = *(const float4*)(feat + (size_t)s * DF + lane * 4);
    float* a = agg + (size_t)d * DF + lane * 4;
    atomicAdd(a + 0, w * f.x);
    atomicAdd(a + 1, w * f.y);
    atomicAdd(a + 2, w * f.z);
    atomicAdd(a + 3, w * f.w);
}

// Block = 256 threads = 8 waves. Block owns M-tile of 16 node rows.
// Wave wv owns output columns [16*wv, 16*wv+16). K = 256 (fused [S*feat | agg]).
__global__ void __launch_bounds__(256)
gemm_kernel(const float* __restrict__ feat,
            const float* __restrict__ agg,
            const float* __restrict__ W,      // [128 x 256] row-major
            const float* __restrict__ bias,   // [128]
            const float* __restrict__ norm,   // [N]
            const float* __restrict__ S,      // [N]
            float* __restrict__ out,          // [N x 128]
            int N) {
    __shared__ float Xs[16 * XS_STRIDE];

    const int tid = threadIdx.x;
    const int m0  = blockIdx.x * 16;

    // Cooperative stage of the fused A-tile: Xs[r][k] = k<128 ? S[row]*feat[row,k]
    //                                                        : agg[row, k-128]
    for (int idx = tid; idx < 16 * 256; idx += 256) {
        const int r   = idx >> 8;     // 0..15
        const int k   = idx & 255;    // 0..255
        const int row = m0 + r;
        float v = 0.0f;
        if (row < N) {
            if (k < DF) v = S[row] * feat[(size_t)row * DF + k];
            else        v = agg[(size_t)row * DF + (k - DF)];
        }
        Xs[r * XS_STRIDE + k] = v;
    }
    __syncthreads();

    const int lane = tid & 31;
    const int wv   = tid >> 5;        // 0..7
    const int nm   = lane & 15;       // M index for A, N index for B/C
    const int half = lane >> 4;       // 0 or 1
    const int ncol = wv * 16 + nm;    // global output column

    const float* __restrict__ wrow = W + (size_t)ncol * 256;
    const float* __restrict__ arow = &Xs[nm * XS_STRIDE];

    v8f c = {};
#pragma unroll
    for (int k0 = 0; k0 < 256; k0 += 4) {
        // A 16x4 fp32: lane m=nm, VGPR j holds K = k0 + 2*half + j
        v2f a;
        a.x = arow[k0 + 2 * half + 0];
        a.y = arow[k0 + 2 * half + 1];
        // B 4x16 fp32: lane n=nm, VGPR j holds K = k0 + 2*half + j; B[k][n] = W[n][k]
        v2f b;
        b.x = wrow[k0 + 2 * half + 0];
        b.y = wrow[k0 + 2 * half + 1];
        c = __builtin_amdgcn_wmma_f32_16x16x4_f32(
                /*neg_a=*/false, a, /*neg_b=*/false, b,
                /*c_mod=*/(short)0, c, /*reuse_a=*/false, /*reuse_b=*/false);
    }

    // C/D layout: VGPR i -> M = i + 8*half, N = nm. Epilogue: norm*(acc + S*b).
#pragma unroll
    for (int i = 0; i < 8; ++i) {
        const int row = m0 + i + 8 * half;
        if (row < N) {
            out[(size_t)row * DF + ncol] = norm[row] * (c[i] + S[row] * bias[ncol]);
        }
    }
}

extern "C" void kernel_launch(void* const* d_in, const int* in_sizes, int n_in,
                              void* d_out, int out_size, void* d_ws, size_t ws_size,
                              hipStream_t stream) {
    const float* feat = (const float*)d_in[0];   // [N,128]
    const float* W    = (const float*)d_in[1];   // [128,256]
    const float* bias = (const float*)d_in[2];   // [128]
    const int*   src  = (const int*)d_in[3];     // [E]
    const int*   dst  = (const int*)d_in[4];     // [E]
    float* out = (float*)d_out;

    const int N = in_sizes[0] / DF;              // 100000
    const int E = in_sizes[3];                   // 500000

    float* ws   = (float*)d_ws;
    float* norm = ws;                 // [N]   (deg, then norm in place)
    float* S    = ws + N;             // [N]
    float* agg  = ws + 2 * (size_t)N; // [N,128]

    const int zcount = N * (2 + DF);
    zero_ws_kernel<<<(zcount + 255) / 256, 256, 0, stream>>>(ws, zcount);
    deg_kernel<<<(E + 255) / 256, 256, 0, stream>>>(dst, norm, E);
    norm_kernel<<<(N + 255) / 256, 256, 0, stream>>>(norm, N);
    scatter_kernel<<<(E + 7) / 8, 256, 0, stream>>>(feat, src, dst, norm, S, agg, E);
    gemm_kernel<<<(N + 15) / 16, 256, 0, stream>>>(feat, agg, W, bias, norm, S, out, N);
}